// BipartiteMPLayer_86260123173231
// MI455X (gfx1250) — compile-verified
//
#include <hip/hip_runtime.h>
#include <math.h>

#define HDIM 128
#define LPITCH 132          // pad LDS rows: stride 132 dwords -> conflict-free B-frag reads
#define LN_EPS 1e-5f

typedef float v2f __attribute__((ext_vector_type(2)));
typedef float v4f __attribute__((ext_vector_type(4)));
typedef float v8f __attribute__((ext_vector_type(8)));

// D = A(16x4 f32) * B(4x16 f32) + C(16x16 f32)
__device__ __forceinline__ v8f wmma_f32(v2f a, v2f b, v8f c) {
  return __builtin_amdgcn_wmma_f32_16x16x4_f32(false, a, false, b, (short)0, c,
                                               false, false);
}

// Cooperatively load the full 128x128 weight matrix into padded LDS.
__device__ __forceinline__ void load_W_lds(const float* __restrict__ W,
                                           float* lds) {
  int tid = threadIdx.x;
#pragma unroll
  for (int base = 0; base < HDIM * HDIM; base += 256 * 4) {
    int idx = base + tid * 4;
    v4f v = *(const v4f*)(W + idx);
    int row = idx >> 7;
    int col = idx & 127;
    *(v4f*)(lds + row * LPITCH + col) = v;
  }
}

// Y[n, o] = sum_h X[n, h] * W[o, h]   (y = x @ W^T), 16 rows per wave.
__global__ __launch_bounds__(256) void gemm_nt(const float* __restrict__ X,
                                               const float* __restrict__ W,
                                               float* __restrict__ Y) {
  __shared__ float w_lds[HDIM * LPITCH];
  load_W_lds(W, w_lds);
  __syncthreads();

  const int wave = threadIdx.x >> 5;
  const int lane = threadIdx.x & 31;
  const int m    = lane & 15;    // row within tile (A) / col within tile (B)
  const int half = lane >> 4;    // K-half selector
  const int n0   = blockIdx.x * 128 + wave * 16;

  // A fragments: A[m, k], k = 4*kc + 2*half + {0,1}
  const float* xrow = X + (size_t)(n0 + m) * HDIM + 2 * half;
  v2f a[32];
#pragma unroll
  for (int kc = 0; kc < 32; ++kc) a[kc] = *(const v2f*)(xrow + 4 * kc);

#pragma unroll
  for (int t = 0; t < 8; ++t) {
    v8f c = {0.f, 0.f, 0.f, 0.f, 0.f, 0.f, 0.f, 0.f};
    const float* bbase = w_lds + (t * 16 + m) * LPITCH + 2 * half;
#pragma unroll
    for (int kc = 0; kc < 32; ++kc) {
      v2f b = *(const v2f*)(bbase + 4 * kc);
      c = wmma_f32(a[kc], b, c);
    }
#pragma unroll
    for (int v = 0; v < 8; ++v)
      Y[(size_t)(n0 + v + 8 * half) * HDIM + t * 16 + m] = c[v];
  }
}

// Out = LayerNorm(X + relu(X@W^T + agg/max(cnt,1) + bias)) * gamma + beta
__global__ __launch_bounds__(256) void gemm_fused_ln(
    const float* __restrict__ X, const float* __restrict__ W,
    const float* __restrict__ agg, const float* __restrict__ cnt,
    const float* __restrict__ bias, const float* __restrict__ gamma,
    const float* __restrict__ beta, float* __restrict__ Out, int Nnode) {
  __shared__ float w_lds[HDIM * LPITCH];
  load_W_lds(W, w_lds);
  __syncthreads();

  const int wave = threadIdx.x >> 5;
  const int lane = threadIdx.x & 31;
  const int m    = lane & 15;
  const int half = lane >> 4;
  const int n0   = blockIdx.x * 128 + wave * 16;

  const float* xrow = X + (size_t)(n0 + m) * HDIM + 2 * half;
  v2f a[32];
#pragma unroll
  for (int kc = 0; kc < 32; ++kc) a[kc] = *(const v2f*)(xrow + 4 * kc);

  v8f acc[8];
#pragma unroll
  for (int t = 0; t < 8; ++t) {
    v8f c = {0.f, 0.f, 0.f, 0.f, 0.f, 0.f, 0.f, 0.f};
    const float* bbase = w_lds + (t * 16 + m) * LPITCH + 2 * half;
#pragma unroll
    for (int kc = 0; kc < 32; ++kc) {
      v2f b = *(const v2f*)(bbase + 4 * kc);
      c = wmma_f32(a[kc], b, c);
    }
    acc[t] = c;
  }

  // Rows owned by this lane (one per C-VGPR index v): row = n0 + v + 8*half.
  int rbase[8];
  float rcnt[8];
#pragma unroll
  for (int v = 0; v < 8; ++v) {
    int row = n0 + v + 8 * half;
    rbase[v] = row;
    rcnt[v] = 1.0f / fmaxf(cnt[row % Nnode], 1.0f);
  }

  // z = x + relu(acc + agg/cnt + b); accumulate row sums for LN.
  float sum[8], sq[8];
#pragma unroll
  for (int v = 0; v < 8; ++v) { sum[v] = 0.f; sq[v] = 0.f; }
#pragma unroll
  for (int t = 0; t < 8; ++t) {
    const int col = t * 16 + m;
    const float bv = bias[col];
#pragma unroll
    for (int v = 0; v < 8; ++v) {
      size_t off = (size_t)rbase[v] * HDIM + col;
      float pre = acc[t][v] + agg[off] * rcnt[v] + bv;
      pre = fmaxf(pre, 0.0f);
      float z = X[off] + pre;
      acc[t][v] = z;  // reuse accumulator registers to hold z
      sum[v] += z;
      sq[v]  += z * z;
    }
  }

  // Reduce across the 16 lanes of this half (rows differ between halves).
#pragma unroll
  for (int v = 0; v < 8; ++v) {
#pragma unroll
    for (int s = 1; s < 16; s <<= 1) {
      sum[v] += __shfl_xor(sum[v], s, 32);
      sq[v]  += __shfl_xor(sq[v], s, 32);
    }
    float mean = sum[v] * (1.0f / HDIM);
    float var  = sq[v] * (1.0f / HDIM) - mean * mean;
    sum[v] = mean;
    sq[v]  = rsqrtf(var + LN_EPS);
  }

#pragma unroll
  for (int t = 0; t < 8; ++t) {
    const int col = t * 16 + m;
    const float gv = gamma[col];
    const float be = beta[col];
#pragma unroll
    for (int v = 0; v < 8; ++v)
      Out[(size_t)rbase[v] * HDIM + col] =
          (acc[t][v] - sum[v]) * sq[v] * gv + be;
  }
}

// agg[b, dst[e], :] += w(e) * Y[b, src[e], :]; 32 lanes per edge, float4 each.
__global__ __launch_bounds__(256) void scatter_add(
    const float* __restrict__ Y, const int* __restrict__ src,
    const int* __restrict__ dst, const float* __restrict__ errw,
    float* __restrict__ agg, int B, int Nsrc, int Ndst, int E) {
  long long gid = (long long)blockIdx.x * blockDim.x + threadIdx.x;
  int e = (int)(gid >> 5);
  int h = ((int)gid & 31) * 4;
  if (e >= E) return;
  int s = src[e];
  int d = dst[e];
  float w = 1.0f;
  if (errw) w = 1.0f / (1.0f + expf(-errw[d]));
  const float* yp = Y + (size_t)s * HDIM + h;
  float* ap = agg + (size_t)d * HDIM + h;
  for (int b = 0; b < B; ++b) {
    v4f val = *(const v4f*)(yp + (size_t)b * Nsrc * HDIM);
    float* o = ap + (size_t)b * Ndst * HDIM;
    atomicAdd(o + 0, val[0] * w);
    atomicAdd(o + 1, val[1] * w);
    atomicAdd(o + 2, val[2] * w);
    atomicAdd(o + 3, val[3] * w);
  }
}

__global__ __launch_bounds__(256) void count_kernel(
    const int* __restrict__ e_dst, float* __restrict__ cnt_e, int E1,
    const int* __restrict__ d_dst, float* __restrict__ cnt_d, int E2) {
  int i = blockIdx.x * blockDim.x + threadIdx.x;
  if (i < E1) atomicAdd(&cnt_e[e_dst[i]], 1.0f);
  if (i < E2) atomicAdd(&cnt_d[d_dst[i]], 1.0f);
}

__global__ __launch_bounds__(256) void zero_kernel(float* __restrict__ p,
                                                   long long n4) {
  long long i = (long long)blockIdx.x * blockDim.x + threadIdx.x;
  if (i < n4) ((v4f*)p)[i] = (v4f){0.f, 0.f, 0.f, 0.f};
}

extern "C" void kernel_launch(void* const* d_in, const int* in_sizes, int n_in,
                              void* d_out, int out_size, void* d_ws,
                              size_t ws_size, hipStream_t stream) {
  const float* hD       = (const float*)d_in[0];
  const float* hE       = (const float*)d_in[1];
  const int*   edge_d2e = (const int*)d_in[2];
  const int*   edge_e2d = (const int*)d_in[3];
  const float* errw     = (const float*)d_in[4];
  const float* W_d2e    = (const float*)d_in[5];
  const float* W_e_self = (const float*)d_in[6];
  const float* b_e      = (const float*)d_in[7];
  const float* g_e      = (const float*)d_in[8];
  const float* beta_e   = (const float*)d_in[9];
  const float* W_e2d    = (const float*)d_in[10];
  const float* W_d_self = (const float*)d_in[11];
  const float* b_d      = (const float*)d_in[12];
  const float* g_d      = (const float*)d_in[13];
  const float* beta_d   = (const float*)d_in[14];

  const int Ne = in_sizes[4];
  const int E  = in_sizes[2] / 2;
  const long long szD = in_sizes[0];            // B*N_d*H floats
  const long long szE = in_sizes[1];            // B*N_e*H floats
  const int B  = (int)(szE / ((long long)Ne * HDIM));
  const int Nd = (int)(szD / ((long long)B * HDIM));
  const long long rowsD = (long long)B * Nd;
  const long long rowsE = (long long)B * Ne;

  float* Y     = (float*)d_ws;       // szE floats (covers both phases)
  float* AGG   = Y + szE;            // szE floats
  float* cnt_e = AGG + szE;          // Ne floats
  float* cnt_d = cnt_e + Ne;         // Nd floats

  float* outD = (float*)d_out;       // hD_new first (tuple order)
  float* outE = outD + szD;          // then hE_new

  // 1) zero AGG + both count arrays (contiguous)
  {
    long long n4 = (szE + Ne + Nd) / 4;
    zero_kernel<<<dim3((unsigned)((n4 + 255) / 256)), dim3(256), 0, stream>>>(
        AGG, n4);
  }
  // 2) degree counts for both directions
  count_kernel<<<dim3((unsigned)((E + 255) / 256)), dim3(256), 0, stream>>>(
      edge_d2e + E, cnt_e, E, edge_e2d + E, cnt_d, E);
  // 3) Y = hD @ W_d2e^T
  gemm_nt<<<dim3((unsigned)(rowsD / 128)), dim3(256), 0, stream>>>(hD, W_d2e, Y);
  // 4) scatter D->E with sigmoid edge weights
  {
    long long thr = (long long)E * 32;
    scatter_add<<<dim3((unsigned)((thr + 255) / 256)), dim3(256), 0, stream>>>(
        Y, edge_d2e, edge_d2e + E, errw, AGG, B, Nd, Ne, E);
  }
  // 5) hE_new = LN(hE + relu(hE@W_e_self^T + AGG/cnt + b_e))
  gemm_fused_ln<<<dim3((unsigned)(rowsE / 128)), dim3(256), 0, stream>>>(
      hE, W_e_self, AGG, cnt_e, b_e, g_e, beta_e, outE, Ne);
  // 6) re-zero AGG for D side
  zero_kernel<<<dim3((unsigned)((szD / 4 + 255) / 256)), dim3(256), 0, stream>>>(
      AGG, szD / 4);
  // 7) Y = hE_new @ W_e2d^T
  gemm_nt<<<dim3((unsigned)(rowsE / 128)), dim3(256), 0, stream>>>(outE, W_e2d, Y);
  // 8) scatter E->D (no edge weights)
  {
    long long thr = (long long)E * 32;
    scatter_add<<<dim3((unsigned)((thr + 255) / 256)), dim3(256), 0, stream>>>(
        Y, edge_e2d, edge_e2d + E, nullptr, AGG, B, Ne, Nd, E);
  }
  // 9) hD_new = LN(hD + relu(hD@W_d_self^T + AGG/cnt + b_d))
  gemm_fused_ln<<<dim3((unsigned)(rowsD / 128)), dim3(256), 0, stream>>>(
      hD, W_d_self, AGG, cnt_d, b_d, g_d, beta_d, outD, Nd);
}